// MsMPN_83949430767932
// MI455X (gfx1250) — compile-verified
//
#include <hip/hip_runtime.h>
#include <hip/hip_bf16.h>
#include <stdint.h>

typedef float v2f __attribute__((ext_vector_type(2)));
typedef float v8f __attribute__((ext_vector_type(8)));
typedef unsigned int v4u __attribute__((ext_vector_type(4)));
typedef int v8i __attribute__((ext_vector_type(8)));
typedef int v4i __attribute__((ext_vector_type(4)));

#define NN   16384
#define DD   128
#define EE   524288
#define NLAY 3

#if __has_builtin(__builtin_amdgcn_tensor_load_to_lds) && __has_builtin(__builtin_amdgcn_s_wait_tensorcnt)
#define USE_TDM 1
#else
#define USE_TDM 0
#endif

// ---------------------------------------------------------------------------
// init: pre0 = X ; out[:, 0:128] = 2*X   (out row stride = 512)
// ---------------------------------------------------------------------------
__global__ __launch_bounds__(256) void mp_init(const float* __restrict__ X,
                                               float* __restrict__ pre0,
                                               float* __restrict__ out) {
    int i = blockIdx.x * 256 + threadIdx.x;          // i < N*D
    int n = i >> 7;
    int d = i & 127;
    float x = X[i];
    pre0[i] = x;
    out[(size_t)n * 512 + d] = 2.0f * x;
}

// ---------------------------------------------------------------------------
// zero scratch (allagg + subagg, contiguous)
// ---------------------------------------------------------------------------
__global__ __launch_bounds__(256) void mp_zero(float* __restrict__ p) {
    p[blockIdx.x * 256 + threadIdx.x] = 0.0f;
}

// ---------------------------------------------------------------------------
// edge scatter: subagg[row] += val * pre[col]   (one block per edge, d=tid)
// sub_indices is int64 per the reference (shape [2,E], rows then cols)
// ---------------------------------------------------------------------------
__global__ __launch_bounds__(128) void mp_scatter(const int64_t* __restrict__ rows,
                                                  const int64_t* __restrict__ cols,
                                                  const float* __restrict__ vals,
                                                  const float* __restrict__ pre,
                                                  float* __restrict__ subagg) {
    int e = blockIdx.x;
    int d = threadIdx.x;
    int r = (int)rows[e];
    int c = (int)cols[e];
    float v = vals[e];
    atomicAdd(&subagg[(size_t)r * DD + d], v * pre[(size_t)c * DD + d]);
}

// ---------------------------------------------------------------------------
// Big GEMM: allagg += A[128-row stripe, Kchunk] @ pre[Kchunk, 128]
// grid = (N/128, KSPLIT); 256 threads = 8 waves; each wave: 16 rows x 128 cols
// A tile staged by the Tensor Data Mover (double-buffered, padded stride 34);
// B tile staged k-inner-4 so WMMA fragments are single ds_load_b64.
// ---------------------------------------------------------------------------
#define KSPLIT 4
#define TK 32
#define ASTRIDE 34   // 32 data + 2 pad DWORDs per row (TDM pad_interval=32, pad_amount=2)

#if USE_TDM
__device__ __forceinline__ void tdm_load_A(const float* gsrc, uint32_t lds_byte_addr) {
    uint64_t ga = (uint64_t)(uintptr_t)gsrc;
    v4u g0;
    g0.x = 1u;                                   // count=1, user descriptor
    g0.y = lds_byte_addr;                        // lds_addr [63:32]
    g0.z = (uint32_t)ga;                         // global_addr low
    g0.w = (uint32_t)(ga >> 32) | (2u << 30);    // global_addr high | type=2
    v8i g1;
    g1[0] = (int)((2u << 16)        // data_size = 2 -> 4 bytes
                | (1u << 20)        // pad_enable
                | (4u << 22)        // pad_interval: 32 DWORDs
                | (1u << 25));      // pad_amount: 2 DWORDs  -> row stride 34 floats
    g1[1] = (int)(((uint32_t)(NN & 0xffff)) << 16);   // tensor_dim0[15:0]
    g1[2] = (int)((((uint32_t)NN >> 16) & 0xffff)     // tensor_dim0[31:16]
                | (((uint32_t)(NN & 0xffff)) << 16)); // tensor_dim1[15:0]
    g1[3] = (int)(((((uint32_t)NN) >> 16) & 0xffff)   // tensor_dim1[31:16]
                | ((uint32_t)TK << 16));              // tile_dim0 = 32
    g1[4] = 128;                                      // tile_dim1 = 128 rows
    g1[5] = NN;                                       // tensor_dim0_stride low32
    g1[6] = 0;
    g1[7] = 0;
    v4i gz4 = {0, 0, 0, 0};
    v8i gz8 = {0, 0, 0, 0, 0, 0, 0, 0};
    // 6-arg form (this toolchain): (g0, g1, g2, g3, g4, cpol)
    __builtin_amdgcn_tensor_load_to_lds(g0, g1, gz4, gz4, gz8, 0);
}
#endif

__global__ __launch_bounds__(256) void mp_gemm_wmma(const float* __restrict__ A,
                                                    const float* __restrict__ B,
                                                    float* __restrict__ C) {
    __shared__ float As[2][128 * ASTRIDE];       // padded A tiles (double buffer)
    __shared__ float Bs[TK / 4][128][4];         // k-inner-4 B tile

    const int t    = threadIdx.x;
    const int wave = t >> 5;            // 0..7
    const int lane = t & 31;
    const int lm   = lane & 15;         // m (A) / n (B) within 16
    const int kh   = (lane >> 4) * 2;   // 0 or 2 : k sub-pair
    const int rowBase = blockIdx.x * 128;
    const int kStart  = blockIdx.y * (NN / KSPLIT);
    const int nTiles  = (NN / KSPLIT) / TK;
    const int m0      = wave * 16;

    v8f acc[8];
#pragma unroll
    for (int i = 0; i < 8; ++i) acc[i] = (v8f)(0.0f);

#if USE_TDM
    if (wave == 0 && lane == 0) {
        tdm_load_A(&A[(size_t)rowBase * NN + kStart], (uint32_t)(uintptr_t)&As[0][0]);
    }
#endif

    for (int tt = 0; tt < nTiles; ++tt) {
        const int k0  = kStart + tt * TK;
        const int buf = tt & 1;

#if !USE_TDM
        // fallback: manual A staging into the padded layout
#pragma unroll
        for (int i = 0; i < 4; ++i) {
            int idx = t + i * 256;                  // 0..1023
            int r   = idx >> 3;
            int c4  = (idx & 7) * 4;
            const float4 va = *(const float4*)&A[(size_t)(rowBase + r) * NN + k0 + c4];
            As[buf][r * ASTRIDE + c4 + 0] = va.x;
            As[buf][r * ASTRIDE + c4 + 1] = va.y;
            As[buf][r * ASTRIDE + c4 + 2] = va.z;
            As[buf][r * ASTRIDE + c4 + 3] = va.w;
        }
#endif
        // ---- stage B tile: 32 rows x 128 cols, k-inner-4 interleave
#pragma unroll
        for (int i = 0; i < 4; ++i) {
            int idx = t + i * 256;
            int r   = idx >> 5;                     // k row 0..31
            int c   = idx & 31;                     // float4 col group
            const float4 vb = *(const float4*)&B[(size_t)(k0 + r) * DD + c * 4];
            Bs[r >> 2][c * 4 + 0][r & 3] = vb.x;
            Bs[r >> 2][c * 4 + 1][r & 3] = vb.y;
            Bs[r >> 2][c * 4 + 2][r & 3] = vb.z;
            Bs[r >> 2][c * 4 + 3][r & 3] = vb.w;
        }
        // prefetch next B tile (global_prefetch_b8)
        if (tt + 1 < nTiles)
            __builtin_prefetch(&B[(size_t)(k0 + TK + (t >> 3)) * DD + (t & 7) * 16], 0, 1);

#if USE_TDM
        if (wave == 0) {
            if (tt + 1 < nTiles) {
                if (lane == 0)
                    tdm_load_A(&A[(size_t)rowBase * NN + k0 + TK],
                               (uint32_t)(uintptr_t)&As[buf ^ 1][0]);
                __builtin_amdgcn_s_wait_tensorcnt(1);   // tile tt complete (in-order)
            } else {
                __builtin_amdgcn_s_wait_tensorcnt(0);
            }
        }
#endif
        __syncthreads();

        const float* __restrict__ Ab = &As[buf][0];
#pragma unroll
        for (int kk = 0; kk < TK; kk += 4) {
            const v2f a = *(const v2f*)&Ab[(m0 + lm) * ASTRIDE + kk + kh];
#pragma unroll
            for (int nb = 0; nb < 8; ++nb) {
                const v2f b = *(const v2f*)&Bs[kk >> 2][nb * 16 + lm][kh];
                acc[nb] = __builtin_amdgcn_wmma_f32_16x16x4_f32(
                    false, a, false, b, (short)0, acc[nb], false, false);
            }
        }
        __syncthreads();
    }

    // ---- accumulate into C (split-K partial sums via f32 atomics)
    const int mOff = (lane >> 4) * 8;
#pragma unroll
    for (int nb = 0; nb < 8; ++nb) {
#pragma unroll
        for (int v = 0; v < 8; ++v) {
            int m = m0 + mOff + v;
            atomicAdd(&C[(size_t)(rowBase + m) * DD + nb * 16 + lm], acc[nb][v]);
        }
    }
}

// ---------------------------------------------------------------------------
// fused layer tail: all_emb/sub_emb -> two DxD GEMMs -> activations ->
// emb (next pre) and l2-normalized emb into output column block.
// ---------------------------------------------------------------------------
__global__ __launch_bounds__(128) void mp_fuse(const float* __restrict__ pre,
                                               const float* __restrict__ allagg,
                                               const float* __restrict__ subagg,
                                               const float* __restrict__ W2,
                                               const float* __restrict__ b2,
                                               const float* __restrict__ W3,
                                               const float* __restrict__ b3,
                                               float* __restrict__ preOut,
                                               float* __restrict__ outBlk) {
    __shared__ float ae[128];
    __shared__ float se[128];
    __shared__ float red[128];

    const int n = blockIdx.x;
    const int o = threadIdx.x;
    const size_t base = (size_t)n * DD;

    float p = pre[base + o];
    ae[o] = p + allagg[base + o];
    se[o] = p * subagg[base + o];
    __syncthreads();

    float h2 = b2[o];
    float h3 = b3[o];
    const float* w2 = &W2[(size_t)o * DD];
    const float* w3 = &W3[(size_t)o * DD];
#pragma unroll 8
    for (int d = 0; d < DD; ++d) {
        h2 = fmaf(ae[d], w2[d], h2);
        h3 = fmaf(se[d], w3[d], h3);
    }
    float sig = 1.0f / (1.0f + __expf(-h2));               // ALPHA = 1
    float lre = (h3 > 0.0f) ? h3 : 0.01f * h3;             // BETA = 1, slope .01
    float e = sig + lre;

    preOut[base + o] = e;

    red[o] = e * e;
    __syncthreads();
#pragma unroll
    for (int s = 64; s > 0; s >>= 1) {
        if (o < s) red[o] += red[o + s];
        __syncthreads();
    }
    float nrm = fmaxf(sqrtf(red[0]), 1e-12f);
    outBlk[(size_t)n * 512 + o] = e / nrm;
}

// ---------------------------------------------------------------------------
// launcher
// ---------------------------------------------------------------------------
extern "C" void kernel_launch(void* const* d_in, const int* in_sizes, int n_in,
                              void* d_out, int out_size, void* d_ws, size_t ws_size,
                              hipStream_t stream) {
    const float*   A    = (const float*)d_in[0];
    const int64_t* sidx = (const int64_t*)d_in[1];   // [2, E] int64
    const float*   sval = (const float*)d_in[2];
    const float*   X    = (const float*)d_in[3];
    const float*   W2   = (const float*)d_in[4];     // [3,128,128]
    const float*   b2   = (const float*)d_in[5];
    const float*   W3   = (const float*)d_in[6];
    const float*   b3   = (const float*)d_in[7];
    float* out = (float*)d_out;                      // [N, 512]

    const size_t ND = (size_t)NN * DD;
    float* pre0   = (float*)d_ws;
    float* pre1   = pre0 + ND;
    float* allagg = pre1 + ND;
    float* subagg = allagg + ND;

    const int64_t* rows = sidx;
    const int64_t* cols = sidx + EE;

    mp_init<<<(unsigned)(ND / 256), 256, 0, stream>>>(X, pre0, out);

    for (int k = 0; k < NLAY; ++k) {
        float* pin  = (k & 1) ? pre1 : pre0;
        float* pout = (k & 1) ? pre0 : pre1;

        // zero allagg + subagg (contiguous)
        mp_zero<<<(unsigned)(2 * ND / 256), 256, 0, stream>>>(allagg);

        mp_scatter<<<EE, 128, 0, stream>>>(rows, cols, sval, pin, subagg);

        dim3 ggrid(NN / 128, KSPLIT);
        mp_gemm_wmma<<<ggrid, 256, 0, stream>>>(A, pin, allagg);

        mp_fuse<<<NN, 128, 0, stream>>>(pin, allagg, subagg,
                                        W2 + (size_t)k * DD * DD, b2 + (size_t)k * DD,
                                        W3 + (size_t)k * DD * DD, b3 + (size_t)k * DD,
                                        pout, out + (size_t)(k + 1) * DD);
    }
}